// Choi_78142634983427
// MI455X (gfx1250) — compile-verified
//
#include <hip/hip_runtime.h>
#include <math.h>

// ---------------------------------------------------------------------------
// Choi matrices of a noisy driven qubit, B=512 batches, M=1024 steps.
//   H(b,m) = px*kron(I,X) + py*kron(I,Y) + (drift+nz)*kron(I,Z)
//   U = expm(-i H dt) = kron(I2, u),  u in SU(2): u = [[a, b], [-b*, a*]]
//   Ucum[m] = U[m]...U[0]  (per-batch time-ordered scan)
//   choi = w w^H,  w = [alpha, -conj(beta), beta, conj(alpha)]
//
// Roofline: ~70 MFLOP vs 70 MB traffic (64 MB stores) -> bandwidth/latency
// bound at fp32 (reference is complex64). Design:
//  * 1 wave32 per batch; lane owns a contiguous 32-step chunk; shfl_up
//    Hillis-Steele wave scan over 4-float SU(2) states.
//  * TDM (tensor_load_to_lds) stages inputs to LDS with hardware padding:
//    1 dword pad per 32 dwords => idx(m) = (m/32)*33 + m%32, making the
//    stride-33 per-lane chunk reads bank-conflict free. One DMA per array.
//  * Choi outer products via V_WMMA_F32_16X16X4_F32: complex 4x4 w w^H equals
//    real 8x8 R(w)(8x2) @ R(w)^T(2x8); two time-steps packed block-diagonally
//    per WMMA; the A/B lane layouts coincide => same VGPR pair for A and B.
//  * Operands for all 16 WMMAs are preloaded into registers first so the 16
//    ds_load_b64 can be claused and waited once, instead of one LDS
//    round-trip stall per WMMA.
// ---------------------------------------------------------------------------

#define M_STEPS 1024

typedef float v2f __attribute__((ext_vector_type(2)));
typedef float v8f __attribute__((ext_vector_type(8)));
typedef unsigned int u32x4 __attribute__((ext_vector_type(4)));
typedef int i32x4 __attribute__((ext_vector_type(4)));
typedef int i32x8 __attribute__((ext_vector_type(8)));

#if defined(__AMDGCN__) && __has_builtin(__builtin_amdgcn_wmma_f32_16x16x4_f32)
#define USE_WMMA 1
#else
#define USE_WMMA 0
#endif

#if defined(__AMDGCN__) && __has_builtin(__builtin_amdgcn_tensor_load_to_lds) && \
    __has_builtin(__builtin_amdgcn_s_wait_tensorcnt)
#define USE_TDM 1
#else
#define USE_TDM 0
#endif

// prev state (ar,ai,br,bi) <- cur (car,cai,cbr,cbi) @ prev   (SU(2) compose)
// alpha' = ca*pa - cb*conj(pb);  beta' = ca*pb + cb*conj(pa)
__device__ __forceinline__ void su2_mul(float car, float cai, float cbr, float cbi,
                                        float& ar, float& ai, float& br, float& bi) {
  float nar = car*ar - cai*ai - (cbr*br + cbi*bi);
  float nai = car*ai + cai*ar - (cbi*br - cbr*bi);
  float nbr = car*br - cai*bi + (cbr*ar + cbi*ai);
  float nbi = car*bi + cai*br + (cbi*ar - cbr*ai);
  ar = nar; ai = nai; br = nbr; bi = nbi;
}

// one time step: P <- u(nz,px,py) @ P
__device__ __forceinline__ void step(float nz, float px, float py, float driftc,
                                     float dt, float& ar, float& ai, float& br, float& bi) {
  float cz = driftc + nz;
  float r2 = px*px + py*py + cz*cz;
  float r  = sqrtf(r2);
  float sn, cs;
  __sincosf(r * dt, &sn, &cs);
  float k = (r > 1e-12f) ? (sn / r) : dt;       // sin(r dt)/r -> dt as r->0
  su2_mul(cs, -k*cz, -k*py, -k*px, ar, ai, br, bi);
}

#if USE_TDM
// One TDM descriptor: linear 1D copy of `nelem` elements (element size per
// dsz_code: 2=4B, 3=8B) into LDS, inserting pad_amt(+1) dwords after every
// 2<<pad_int dwords.  D# packing per CDNA5 ISA 8.3-8.6.
__device__ __forceinline__ void tdm_load_1d(unsigned lds_off, const void* gptr,
                                            unsigned nelem, unsigned dsz_code,
                                            unsigned pad_int, unsigned pad_amt) {
  unsigned long long ga = (unsigned long long)(uintptr_t)gptr;
  u32x4 g0 = {0u, 0u, 0u, 0u};
  g0[0] = 1u;                                        // count=1 (valid), load
  g0[1] = lds_off;                                   // lds_addr (bytes)
  g0[2] = (unsigned)ga;                              // global_addr[31:0]
  g0[3] = (unsigned)((ga >> 32) & 0x01FFFFFFu) | (2u << 30);  // addr hi, type=2
  i32x8 g1 = {0, 0, 0, 0, 0, 0, 0, 0};
  g1[0] = (int)((dsz_code << 16) | (1u << 20) |      // data_size, pad_enable
                (pad_int << 22) | (pad_amt << 25));  // pad interval/amount
  g1[1] = (int)((nelem & 0xFFFFu) << 16);            // tensor_dim0[15:0]
  g1[2] = (int)(((nelem >> 16) & 0xFFFFu) | (1u << 16)); // dim0 hi, tensor_dim1=1
  g1[3] = (int)((nelem & 0xFFFFu) << 16);            // tile_dim0 = nelem
  g1[4] = 1;                                         // tile_dim1=1, tile_dim2=0
  g1[5] = (int)nelem;                                // tensor_dim0_stride
  i32x4 gz = {0, 0, 0, 0};
#if defined(__clang_major__) && (__clang_major__ >= 23)
  i32x8 gz8 = {0, 0, 0, 0, 0, 0, 0, 0};
  __builtin_amdgcn_tensor_load_to_lds(g0, g1, gz, gz, gz8, 0);
#else
  __builtin_amdgcn_tensor_load_to_lds(g0, g1, gz, gz, 0);
#endif
}
#endif

__global__ __launch_bounds__(32)
void choi_scan_kernel(const float* __restrict__ noise,   // (B, M)
                      const float* __restrict__ pulses,  // (B, M, 2)
                      const float* __restrict__ drift,   // complex64, [0]=re
                      float* __restrict__ out)           // (B, M, 4, 4) c64
{
  const int b    = blockIdx.x;
  const int lane = threadIdx.x;
  const float dt = 1.0f / (float)M_STEPS;

  // padded-linear layouts: nz element m at (m/32)*33 + m%32 ;
  // pulse pair m at float offset (m/32)*66 + (m%32)*2
  __shared__ float s_nz [32 * 33];
  __shared__ float s_pxy[32 * 66];
  __shared__ float s_w[32][8];      // per-lane w: {wr0,wi0,...,wr3,wi3}

  const float driftc = drift[0];
  const size_t base = (size_t)b * M_STEPS;

#if USE_TDM
  // DMA both input blocks into LDS with hardware row padding.
  tdm_load_1d((unsigned)(uintptr_t)(void*)s_nz,  noise + base,      1024u,
              /*4B*/2u, /*pad per 32dw*/4u, /*1 dw*/0u);
  tdm_load_1d((unsigned)(uintptr_t)(void*)s_pxy, pulses + 2 * base, 1024u,
              /*8B*/3u, /*pad per 64dw*/5u, /*2 dw*/1u);
  __builtin_amdgcn_s_wait_tensorcnt(0);
  __syncthreads();
#else
  for (int c = 0; c < 32; ++c) {
    int m = c * 32 + lane;                               // coalesced global
    s_nz[c * 33 + lane] = noise[base + m];
    float2 pu = reinterpret_cast<const float2*>(pulses)[base + m];
    *reinterpret_cast<float2*>(&s_pxy[c * 66 + 2 * lane]) = pu;
  }
  __syncthreads();
#endif

  // ---- pass 1: per-lane chunk product over m = lane*32 .. lane*32+31
  float ar = 1.f, ai = 0.f, br = 0.f, bi = 0.f;
  for (int s = 0; s < 32; ++s) {
    float2 pq = *reinterpret_cast<const float2*>(&s_pxy[lane * 66 + 2 * s]);
    step(s_nz[lane * 33 + s], pq.x, pq.y, driftc, dt, ar, ai, br, bi);
  }

  // ---- inclusive wave scan (later @ earlier), wave32 shuffles
  for (int k = 1; k < 32; k <<= 1) {
    float qar = __shfl_up(ar, k), qai = __shfl_up(ai, k);
    float qbr = __shfl_up(br, k), qbi = __shfl_up(bi, k);
    if (lane >= k) {
      float tar = qar, tai = qai, tbr = qbr, tbi = qbi;
      su2_mul(ar, ai, br, bi, tar, tai, tbr, tbi);   // S = S_cur @ S_{lane-k}
      ar = tar; ai = tai; br = tbr; bi = tbi;
    }
  }
  // exclusive prefix
  {
    float ear = __shfl_up(ar, 1), eai = __shfl_up(ai, 1);
    float ebr = __shfl_up(br, 1), ebi = __shfl_up(bi, 1);
    if (lane == 0) { ear = 1.f; eai = 0.f; ebr = 0.f; ebi = 0.f; }
    ar = ear; ai = eai; br = ebr; bi = ebi;
  }

#if USE_WMMA
  // s-invariant WMMA roles.  A(16x4) f32: VGPR0 = K0/K2 lane halves, VGPR1 =
  // K1/K3; rows 0-7 cols 0-1 = R(w_a), rows 8-15 cols 2-3 = R(w_b); B(4x16)
  // mirrors it so A and B are the same VGPR pair.
  const bool valid  = (lane < 8) || (lane >= 24);
  const int  srcOff = (lane < 8) ? 0 : 1;               // lane 2p vs 2p+1
  const int  row    = lane & 7;
  const int  ri     = row >> 1;
  const bool even   = ((row & 1) == 0);
  const bool isA    = (lane < 8)  && ((lane & 1) == 0); // holds choi_a col j
  const bool isB    = (lane >= 24) && ((lane & 1) == 0);// holds choi_b col j
  const int  jcol   = (lane & 7) >> 1;
#endif

  // ---- pass 2: rebuild per-step prefixes, emit choi = w w^H
#pragma unroll 1
  for (int s = 0; s < 32; ++s) {
    float2 pq = *reinterpret_cast<const float2*>(&s_pxy[lane * 66 + 2 * s]);
    step(s_nz[lane * 33 + s], pq.x, pq.y, driftc, dt, ar, ai, br, bi);
    // w = [alpha, -conj(beta), beta, conj(alpha)]
    // wr = {ar,-br, br, ar}, wi = {ai, bi, bi,-ai}

#if USE_WMMA
    s_w[lane][0] = ar;  s_w[lane][1] = ai;
    s_w[lane][2] = -br; s_w[lane][3] = bi;
    s_w[lane][4] = br;  s_w[lane][5] = bi;
    s_w[lane][6] = ar;  s_w[lane][7] = -ai;
    __syncthreads();

    // Preload operands for all 16 pairs: lets the compiler clause the 16
    // ds_load_b64 and wait once, then stream the WMMAs.
    v2f av[16];
#pragma unroll
    for (int p = 0; p < 16; ++p) {
      float2 wri = *reinterpret_cast<const float2*>(&s_w[2 * p + srcOff][2 * ri]);
      float a0 = even ?  wri.x : wri.y;   // col 0 of R(w): even->wr, odd->wi
      float a1 = even ? -wri.y : wri.x;   // col 1 of R(w): even->-wi, odd->wr
      av[p].x = valid ? a0 : 0.f;
      av[p].y = valid ? a1 : 0.f;
    }

#pragma unroll
    for (int p = 0; p < 16; ++p) {     // pair (lane 2p, lane 2p+1) per WMMA
      v8f cz = {0.f, 0.f, 0.f, 0.f, 0.f, 0.f, 0.f, 0.f};
      v8f acc = __builtin_amdgcn_wmma_f32_16x16x4_f32(
          false, av[p], false, av[p], (short)0, cz, false, false);
      // lane 2j holds column j of choi_a as (re,im) pairs in acc[0..7];
      // lane 24+2j holds column j of choi_b likewise.
      if (isA || isB) {
        int srcLane = 2 * p + (isB ? 1 : 0);
        size_t m = (size_t)srcLane * 32 + (size_t)s;
        float* o = out + (base + m) * 32 + (size_t)jcol * 2;
#pragma unroll
        for (int i2 = 0; i2 < 4; ++i2) {
          float2 e; e.x = acc[2 * i2]; e.y = acc[2 * i2 + 1];
          *reinterpret_cast<float2*>(o + i2 * 8) = e;   // entry (i2, jcol)
        }
      }
    }
    __syncthreads();
#else
    float wr[4] = { ar, -br, br, ar };
    float wi[4] = { ai,  bi, bi, -ai };
    float* o = out + (base + (size_t)lane * 32 + (size_t)s) * 32;
#pragma unroll
    for (int i = 0; i < 4; ++i) {
      float4 q0, q1;
      q0.x = wr[i]*wr[0] + wi[i]*wi[0];  q0.y = wi[i]*wr[0] - wr[i]*wi[0];
      q0.z = wr[i]*wr[1] + wi[i]*wi[1];  q0.w = wi[i]*wr[1] - wr[i]*wi[1];
      q1.x = wr[i]*wr[2] + wi[i]*wi[2];  q1.y = wi[i]*wr[2] - wr[i]*wi[2];
      q1.z = wr[i]*wr[3] + wi[i]*wi[3];  q1.w = wi[i]*wr[3] - wr[i]*wi[3];
      *reinterpret_cast<float4*>(o + i * 8)     = q0;
      *reinterpret_cast<float4*>(o + i * 8 + 4) = q1;
    }
#endif
  }
}

extern "C" void kernel_launch(void* const* d_in, const int* in_sizes, int n_in,
                              void* d_out, int out_size, void* d_ws, size_t ws_size,
                              hipStream_t stream) {
  (void)n_in; (void)out_size; (void)d_ws; (void)ws_size;
  const float* noise  = (const float*)d_in[0];   // (B,M,1) f32
  const float* pulses = (const float*)d_in[1];   // (B,M,2) f32
  const float* drift  = (const float*)d_in[5];   // (1,) complex64 -> [0]=re
  float* out = (float*)d_out;                    // (B,M,4,4) c64 interleaved
  int B = in_sizes[0] / M_STEPS;
  choi_scan_kernel<<<dim3(B), dim3(32), 0, stream>>>(noise, pulses, drift, out);
}